// VQCAudit_32100585570993
// MI455X (gfx1250) — compile-verified
//
#include <hip/hip_runtime.h>
#include <math.h>

#define NQ 14
#define NL 6
#define NBATCH 128
#define DIM 16384      // 2^14 real amplitudes (state is provably real)
#define TPB 512        // 16 wave32 per workgroup
#define USTRIDE 784    // per-layer: 3 * 16x16 group unitaries + one 4x4

typedef __attribute__((ext_vector_type(2))) float v2f;
typedef __attribute__((ext_vector_type(8))) float v8f;

// LDS address of element (group-row k, column tile*16+n) for group g's axis.
// index i = g0*1024 + g1*64 + g2*4 + g3  (qubit 0 = MSB)
__device__ __forceinline__ int group_addr(int g, int k, int tile, int n) {
  if (g == 0) return (k << 10) + (tile << 4) + n;                                // g0 axis
  if (g == 1) return ((tile >> 2) << 10) + (k << 6) + ((tile & 3) << 4) + n;     // g1 axis
  return (tile << 8) + ((n >> 2) << 6) + (k << 2) + (n & 3);                     // g2 axis
}

__global__ __launch_bounds__(TPB) void vqc_wmma_kernel(
    const float* __restrict__ x, const float* __restrict__ prm,
    float* __restrict__ out) {
  __shared__ float s[DIM];            // real statevector, 64 KB
  __shared__ float u[NL * USTRIDE];   // fused group unitaries
  __shared__ float cs[2 * NL * NQ];   // cos/sin(theta/2) for all layer params
  __shared__ float xc[NQ];
  __shared__ float xs[NQ];
  __shared__ float red[TPB];

  const int t = threadIdx.x;
  const int b = blockIdx.x;
  const int lane = t & 31;
  const int wave = t >> 5;
  const int n = lane & 15;   // N index / M index inside a 16-wide fragment
  const int h = lane >> 4;   // lane half

  // ---- phase 0: trig tables ----
  if (t < NL * NQ) {
    float th = 0.5f * prm[t];
    cs[2 * t] = cosf(th);
    cs[2 * t + 1] = sinf(th);
  }
  if (t < NQ) {
    float th = 0.5f * x[b * NQ + t];
    xc[t] = cosf(th);
    xs[t] = sinf(th);
  }
  __syncthreads();

  // ---- phase 1: build fused group unitaries U = C_odd * C_even * kron(RY) ----
  for (int e = t; e < NL * USTRIDE; e += TPB) {
    int l = e / USTRIDE;
    int r = e - l * USTRIDE;
    const float* lcs = &cs[2 * l * NQ];
    float val = 1.0f;
    if (r < 768) {                       // 16x16 groups g = 0,1,2 (qubits 4g..4g+3)
      int g = r >> 8;
      int ij = r & 255;
      int i = ij >> 4, j = ij & 15;
      // odd in-group CNOT (q+1,q+2): ctrl bit2 -> flip bit1
      int so = i ^ (((i >> 2) & 1) << 1);
      // even CNOTs (q,q+1),(q+2,q+3): ctrl bit3->flip bit2, ctrl bit1->flip bit0
      int rr = so ^ (((so >> 3) & 1) << 2) ^ ((so >> 1) & 1);
      for (int bb = 3; bb >= 0; --bb) {  // bit3 = qubit 4g (MSB of group)
        int q = 4 * g + (3 - bb);
        float c = lcs[2 * q], sn = lcs[2 * q + 1];
        int rb = (rr >> bb) & 1, jb = (j >> bb) & 1;
        val *= jb ? (rb ? c : -sn) : (rb ? sn : c);   // RY = [[c,-s],[s,c]]
      }
    } else {                              // 4x4 group (qubits 12,13) incl CNOT(12,13)
      int ij = r - 768;
      int i = ij >> 2, j = ij & 3;
      int rr = i ^ ((i >> 1) & 1);        // ctrl bit1 -> flip bit0
      for (int bb = 1; bb >= 0; --bb) {
        int q = 12 + (1 - bb);
        float c = lcs[2 * q], sn = lcs[2 * q + 1];
        int rb = (rr >> bb) & 1, jb = (j >> bb) & 1;
        val *= jb ? (rb ? c : -sn) : (rb ? sn : c);
      }
    }
    u[e] = val;
  }

  // ---- phase 2: product-state init (encoding RY layer on |0..0>) ----
  for (int k = 0; k < DIM / TPB; ++k) {
    int i = t * (DIM / TPB) + k;
    float amp = 1.0f;
#pragma unroll
    for (int q = 0; q < NQ; ++q) amp *= ((i >> (13 - q)) & 1) ? xs[q] : xc[q];
    s[i] = amp;
  }
  __syncthreads();

  // ---- phase 3: layers ----
  for (int l = 0; l < NL; ++l) {
    const float* Ul = &u[l * USTRIDE];

    // three 16x16 group applications, each = U(16x16) @ S(16x1024) via WMMA f32 16x16x4
#pragma unroll
    for (int g = 0; g < 3; ++g) {
      const float* Ug = Ul + (g << 8);
      // A fragments: A[m][k], m = lane&15, k = 4*kc + 2*h + v
      v2f a[4];
#pragma unroll
      for (int kc = 0; kc < 4; ++kc) {
        int k0 = 4 * kc + 2 * h;
        v2f av;
        av[0] = Ug[n * 16 + k0];
        av[1] = Ug[n * 16 + k0 + 1];
        a[kc] = av;
      }
      // B fragments for this wave's 4 column tiles (read everything, then barrier)
      float bf[4][4][2];
#pragma unroll
      for (int it = 0; it < 4; ++it) {
        int tile = wave * 4 + it;
#pragma unroll
        for (int kc = 0; kc < 4; ++kc) {
          int k0 = 4 * kc + 2 * h;
          bf[it][kc][0] = s[group_addr(g, k0, tile, n)];
          bf[it][kc][1] = s[group_addr(g, k0 + 1, tile, n)];
        }
      }
      __syncthreads();  // all column fibers read -> in-place store is safe
#pragma unroll
      for (int it = 0; it < 4; ++it) {
        int tile = wave * 4 + it;
        v8f acc = {};
#pragma unroll
        for (int kc = 0; kc < 4; ++kc) {
          v2f bv;
          bv[0] = bf[it][kc][0];
          bv[1] = bf[it][kc][1];
          acc = __builtin_amdgcn_wmma_f32_16x16x4_f32(
              false, a[kc], false, bv, (short)0, acc, false, false);
        }
#pragma unroll
        for (int v = 0; v < 8; ++v)
          s[group_addr(g, v + 8 * h, tile, n)] = acc[v];  // D: m = v + 8*h
      }
      __syncthreads();
    }

    // 4x4 group (qubits 12,13) fused with the 3 cross-group CNOTs
    // (3,4): ctrl bit10 -> flip bit9; (7,8): ctrl bit6 -> flip bit5;
    // (11,12): ctrl bit2 -> flip bit1.  Disjoint -> one XOR permutation.
    const float* U3 = Ul + 768;
    float fin[32];
#pragma unroll
    for (int k = 0; k < 32; ++k) fin[k] = s[t * 32 + k];
    __syncthreads();
#pragma unroll
    for (int r = 0; r < 8; ++r) {
#pragma unroll
      for (int mp = 0; mp < 4; ++mp) {
        float v = U3[mp * 4 + 0] * fin[r * 4 + 0] + U3[mp * 4 + 1] * fin[r * 4 + 1] +
                  U3[mp * 4 + 2] * fin[r * 4 + 2] + U3[mp * 4 + 3] * fin[r * 4 + 3];
        int i = t * 32 + r * 4 + mp;
        int d = i ^ (((i >> 10) & 1) << 9) ^ (((i >> 6) & 1) << 5) ^
                (((i >> 2) & 1) << 1);
        s[d] = v;
      }
    }
    __syncthreads();
  }

  // ---- phase 4: <Z0> = sum_{i<8192} s^2 - sum_{i>=8192} s^2 ----
  float local = 0.0f;
#pragma unroll
  for (int k = 0; k < 32; ++k) {
    float v = s[t * 32 + k];
    local += v * v;
  }
  if (t >= 256) local = -local;  // t*32 >= 8192  <=>  qubit-0 bit set
  red[t] = local;
  __syncthreads();
  for (int off = TPB / 2; off > 0; off >>= 1) {
    if (t < off) red[t] += red[t + off];
    __syncthreads();
  }
  if (t == 0) out[b] = red[0];
}

extern "C" void kernel_launch(void* const* d_in, const int* in_sizes, int n_in,
                              void* d_out, int out_size, void* d_ws, size_t ws_size,
                              hipStream_t stream) {
  (void)in_sizes; (void)n_in; (void)d_ws; (void)ws_size; (void)out_size;
  const float* x = (const float*)d_in[0];     // (128, 14) f32
  const float* prm = (const float*)d_in[1];   // (84,) f32
  float* out = (float*)d_out;                 // (128,) f32
  vqc_wmma_kernel<<<NBATCH, TPB, 0, stream>>>(x, prm, out);
}